// InputLayer_69750268887704
// MI455X (gfx1250) — compile-verified
//
#include <hip/hip_runtime.h>

// Categorical input-layer forward:
//   out[n, b] = log(params[s_pids[n] + data[vids[n], b]])
// Shapes: data[256,512] i32, params[65536*256] f32, vids/s_pids[65536] i32,
// out[65536,512] f32.

namespace {
constexpr int kCats    = 256;
constexpr int kBatch   = 512;
constexpr int kNodes   = 65536;
constexpr int kNPB     = 4;               // nodes staged per tile
constexpr int kThreads = 256;             // 8 x wave32
constexpr int kTiles   = kNodes / kNPB;   // 16384
constexpr int kGrid    = 4096;            // 4 tiles per block (persistent loop)
}

typedef float v2f __attribute__((ext_vector_type(2)));
typedef int   v2i __attribute__((ext_vector_type(2)));

__global__ __launch_bounds__(kThreads, 2)
void cat_logmars_kernel(const int*   __restrict__ data,
                        const float* __restrict__ params,
                        const int*   __restrict__ vids,
                        const int*   __restrict__ spids,
                        float*       __restrict__ out)
{
    // Double-buffered per-node log tables: 2 x 4 nodes x 256 cats x f32 = 8 KiB
    __shared__ float lp[2][kNPB][kCats];

    const int tid = threadIdx.x;
    int buf = 0;

    for (int tile = blockIdx.x; tile < kTiles; tile += gridDim.x, buf ^= 1) {
        const int base = tile * kNPB;

        // ---- Speculative prefetch of the NEXT tile's param rows (4 KiB).
        // One vector prefetch: 256 lanes, 16 B apart -> covers kNPB rows.
        // Emits global_prefetch_b8 (gfx1250); errors are silently dropped.
        const int nxt = tile + gridDim.x;
        if (nxt < kTiles) {
            const int nspid = spids[nxt * kNPB];            // uniform -> s_load
            __builtin_prefetch(&params[nspid + tid * 4], 0, 0);
        }

        // ---- Stage: log of each category, once per node row.
        int vid[kNPB];
        #pragma unroll
        for (int k = 0; k < kNPB; ++k) {
            const int node = base + k;                      // uniform per block
            vid[k] = vids[node];                            // -> s_load
            const int sp = spids[node];                     // -> s_load
            // Row is consumed exactly once into LDS: non-temporal load.
            const float p = __builtin_nontemporal_load(&params[sp + tid]);
            // ln(x) = log2(x) * ln(2); v_log_f32 + v_mul_f32
            lp[buf][k][tid] = 0.69314718055994530942f * __builtin_amdgcn_logf(p);
        }
        __syncthreads();   // staging of `buf` complete; also retires all waves'
                           // gathers from buf^1 one iteration back.

        // ---- Gather: 2 batch elements per thread per node, v2 loads/stores.
        const int b = tid * 2;
        #pragma unroll
        for (int k = 0; k < kNPB; ++k) {
            const v2i c = *(const v2i*)(data + (size_t)vid[k] * kBatch + b);
            v2f o;
            o.x = lp[buf][k][c.x & (kCats - 1)];
            o.y = lp[buf][k][c.y & (kCats - 1)];
            // Output is streaming write-once: non-temporal b64 store.
            __builtin_nontemporal_store(
                o, (v2f*)(out + (size_t)(base + k) * kBatch + b));
        }
        // No trailing barrier: double buffering makes the next stage safe.
    }
}

extern "C" void kernel_launch(void* const* d_in, const int* in_sizes, int n_in,
                              void* d_out, int out_size, void* d_ws, size_t ws_size,
                              hipStream_t stream)
{
    const int*   data   = (const int*)  d_in[0];
    const float* params = (const float*)d_in[1];
    const int*   vids   = (const int*)  d_in[2];
    const int*   spids  = (const int*)  d_in[3];
    float*       out    = (float*)      d_out;

    hipLaunchKernelGGL(cat_logmars_kernel, dim3(kGrid), dim3(kThreads), 0, stream,
                       data, params, vids, spids, out);
}